// GATLayer_86535001079879
// MI455X (gfx1250) — compile-verified
//
#include <hip/hip_runtime.h>

typedef float v2f __attribute__((ext_vector_type(2)));
typedef float v8f __attribute__((ext_vector_type(8)));

#define NN 50000
#define KK 128
#define RR 8
#define NR (NN * RR)
#define H1 512
#define KR 1024
#define EPS 1e-8f

// ---- monotone float<->uint encoding for atomic max ----
__device__ __forceinline__ unsigned enc_f(float f) {
    unsigned u = __float_as_uint(f);
    return (u & 0x80000000u) ? ~u : (u | 0x80000000u);
}
__device__ __forceinline__ float dec_f(unsigned u) {
    return (u & 0x80000000u) ? __uint_as_float(u ^ 0x80000000u) : __uint_as_float(~u);
}

// issue one 8-float strip (two async b128) global -> LDS, tracked by ASYNCcnt
__device__ __forceinline__ void async_copy8(uint32_t lds_addr, uint64_t gaddr) {
    asm volatile(
        "global_load_async_to_lds_b128 %0, %1, off\n\t"
        "global_load_async_to_lds_b128 %0, %1, off offset:16"
        :: "v"(lds_addr), "v"(gaddr) : "memory");
}
__device__ __forceinline__ void wait_async0() {
    asm volatile("s_wait_asynccnt 0" ::: "memory");
}

// ---------------- Phase A: per-edge attention + segment max ----------------
__global__ void __launch_bounds__(256) att_kernel(
    const float* __restrict__ nodes, const float* __restrict__ rels,
    const int* __restrict__ froms, const int* __restrict__ tos,
    const int* __restrict__ relidx, float* __restrict__ att,
    unsigned* __restrict__ rowmax, int E) {
    int e = blockIdx.x * 8 + (threadIdx.x >> 5);
    if (e >= E) return;
    int lane = threadIdx.x & 31;
    int f = froms[e], t = tos[e], r = relidx[e];
    float4 a = *(const float4*)(nodes + (size_t)f * KK + lane * 4);
    float4 b = *(const float4*)(nodes + (size_t)t * KK + lane * 4);
    float4 c = *(const float4*)(rels  + (size_t)r * KK + lane * 4);
    float v = a.x * c.x * b.x + a.y * c.y * b.y + a.z * c.z * b.z + a.w * c.w * b.w;
    #pragma unroll
    for (int off = 16; off > 0; off >>= 1) v += __shfl_xor(v, off, 32);
    if (lane == 0) {
        att[e] = v;
        atomicMax(rowmax + (f + r * NN), enc_f(v));
    }
}

// ---------------- Phase B: exp + segment sum ----------------
__global__ void __launch_bounds__(256) exp_kernel(
    const int* __restrict__ froms, const int* __restrict__ relidx,
    float* __restrict__ att, const unsigned* __restrict__ rowmax,
    float* __restrict__ denom, int E) {
    int e = blockIdx.x * 256 + threadIdx.x;
    if (e >= E) return;
    int row = froms[e] + relidx[e] * NN;
    float ex = expf(att[e] - dec_f(rowmax[row]));
    att[e] = ex;
    atomicAdd(denom + row, ex);
}

// ---------------- Phase C: normalized scatter-add into agg ----------------
__global__ void __launch_bounds__(256) scatter_kernel(
    const float* __restrict__ nodes, const int* __restrict__ froms,
    const int* __restrict__ tos, const int* __restrict__ relidx,
    const float* __restrict__ att, const float* __restrict__ denom,
    float* __restrict__ agg, int E) {
    int e = blockIdx.x * 8 + (threadIdx.x >> 5);
    if (e >= E) return;
    int lane = threadIdx.x & 31;
    int f = froms[e], t = tos[e], r = relidx[e];
    int row = f + r * NN;
    float val = att[e] / denom[row];
    float4 x = *(const float4*)(nodes + (size_t)t * KK + lane * 4);
    float* dst = agg + (size_t)row * KK + lane * 4;
    atomicAdd(dst + 0, val * x.x);
    atomicAdd(dst + 1, val * x.y);
    atomicAdd(dst + 2, val * x.z);
    atomicAdd(dst + 3, val * x.w);
}

// ---------------- GEMM1: h1 = relu(nw @ W1 + b1), nw[m, r*128+k] = agg[(r*N+m)*128+k] ----
// One block = 16 output rows; 8 waves x 4 N-tiles cover all 512 cols.
// A-tiles double-buffered in LDS via async global->LDS copies (ASYNCcnt).
__global__ void __launch_bounds__(256) gemm1_kernel(
    const float* __restrict__ agg, const float* __restrict__ W1,
    const float* __restrict__ b1, float* __restrict__ h1) {
    __shared__ float As[2][16][132];  // +4 pad -> conflict-free ds_load_b64 fragments
    const int tid  = threadIdx.x;
    const int wave = tid >> 5;
    const int lane = tid & 31;
    const int hi   = lane >> 4;    // K-half select
    const int lrow = lane & 15;
    const int m0   = blockIdx.x * 16;
    // per-thread staging strip: 8 consecutive floats
    const int fidx = tid * 8;
    const int ti = fidx >> 7, tj = fidx & 127;

    v8f acc[4] = {};

    // prologue: stage r = 0 into buffer 0
    async_copy8((uint32_t)(uintptr_t)&As[0][ti][tj],
                (uint64_t)(uintptr_t)(agg + ((size_t)(m0 + ti)) * KK + tj));
    wait_async0();
    __syncthreads();

    for (int r = 0; r < RR; ++r) {
        const int buf = r & 1;
        if (r + 1 < RR) {  // overlap next A-tile fetch with this chunk's WMMAs
            async_copy8((uint32_t)(uintptr_t)&As[buf ^ 1][ti][tj],
                        (uint64_t)(uintptr_t)(agg + ((size_t)((r + 1) * NN + m0 + ti)) * KK + tj));
        }
        for (int kk = 0; kk < KK; kk += 4) {
            v2f a = *(const v2f*)(&As[buf][lrow][kk + hi * 2]);
            int kg = r * KK + kk + hi * 2;  // global row of W1
            #pragma unroll
            for (int t = 0; t < 4; ++t) {
                int n0 = (wave + t * 8) * 16;
                v2f b;
                b.x = W1[(size_t)kg * H1 + n0 + lrow];
                b.y = W1[(size_t)(kg + 1) * H1 + n0 + lrow];
                acc[t] = __builtin_amdgcn_wmma_f32_16x16x4_f32(
                    false, a, false, b, (short)0, acc[t], false, false);
            }
        }
        if (r + 1 < RR) {
            wait_async0();     // drain this wave's issues
            __syncthreads();   // tile visible + old buffer free for reuse
        }
    }
    #pragma unroll
    for (int t = 0; t < 4; ++t) {
        int col = (wave + t * 8) * 16 + lrow;
        float bias = b1[col];
        #pragma unroll
        for (int v = 0; v < 8; ++v) {
            int row = m0 + v + hi * 8;
            float x = acc[t][v] + bias;
            h1[(size_t)row * H1 + col] = x > 0.f ? x : 0.f;
        }
    }
}

// ---------------- GEMM2: h2 = h1 @ W2 + b2 -> d_out ----------------
__global__ void __launch_bounds__(256) gemm2_kernel(
    const float* __restrict__ h1, const float* __restrict__ W2,
    const float* __restrict__ b2, float* __restrict__ h2) {
    __shared__ float As[2][16][132];
    const int tid  = threadIdx.x;
    const int wave = tid >> 5;     // tile_n = wave (8 tiles = 128 cols)
    const int lane = tid & 31;
    const int hi   = lane >> 4;
    const int lrow = lane & 15;
    const int m0   = blockIdx.x * 16;
    const int n0   = wave * 16;
    const int fidx = tid * 8;
    const int ti = fidx >> 7, tj = fidx & 127;

    v8f acc = {};

    async_copy8((uint32_t)(uintptr_t)&As[0][ti][tj],
                (uint64_t)(uintptr_t)(h1 + (size_t)(m0 + ti) * H1 + tj));
    wait_async0();
    __syncthreads();

    for (int cc = 0; cc < 4; ++cc) {  // K-chunks of 128 (H1 = 512)
        const int buf = cc & 1;
        if (cc + 1 < 4) {
            async_copy8((uint32_t)(uintptr_t)&As[buf ^ 1][ti][tj],
                        (uint64_t)(uintptr_t)(h1 + (size_t)(m0 + ti) * H1 + (cc + 1) * 128 + tj));
        }
        for (int kk = 0; kk < 128; kk += 4) {
            v2f a = *(const v2f*)(&As[buf][lrow][kk + hi * 2]);
            int kg = cc * 128 + kk + hi * 2;
            v2f b;
            b.x = W2[(size_t)kg * KK + n0 + lrow];
            b.y = W2[(size_t)(kg + 1) * KK + n0 + lrow];
            acc = __builtin_amdgcn_wmma_f32_16x16x4_f32(
                false, a, false, b, (short)0, acc, false, false);
        }
        if (cc + 1 < 4) {
            wait_async0();
            __syncthreads();
        }
    }
    int col = n0 + lrow;
    float bias = b2[col];
    #pragma unroll
    for (int v = 0; v < 8; ++v) {
        int row = m0 + v + hi * 8;
        h2[(size_t)row * KK + col] = acc[v] + bias;
    }
}

// ---------------- batch-norm stats (per column) ----------------
__global__ void __launch_bounds__(256) stats_kernel(
    const float* __restrict__ h2, float* __restrict__ stats) {
    int col = blockIdx.x;  // 0..127
    float s = 0.f, s2 = 0.f;
    for (int i = threadIdx.x; i < NN; i += 256) {
        float x = h2[(size_t)i * KK + col];
        s += x; s2 += x * x;
    }
    __shared__ float sh[256], sh2[256];
    sh[threadIdx.x] = s; sh2[threadIdx.x] = s2;
    __syncthreads();
    for (int off = 128; off > 0; off >>= 1) {
        if (threadIdx.x < off) {
            sh[threadIdx.x]  += sh[threadIdx.x + off];
            sh2[threadIdx.x] += sh2[threadIdx.x + off];
        }
        __syncthreads();
    }
    if (threadIdx.x == 0) {
        float mean = sh[0] / (float)NN;
        float var  = (sh2[0] - sh[0] * mean) / (float)(NN - 1);  // ddof=1
        stats[col]       = mean;
        stats[KK + col]  = rsqrtf(var + EPS);
    }
}

// ---------------- normalize (in place on d_out) ----------------
__global__ void __launch_bounds__(256) norm_kernel(
    float* __restrict__ h2, const float* __restrict__ stats,
    const float* __restrict__ normp) {
    int idx = blockIdx.x * 256 + threadIdx.x;
    if (idx >= NN * KK) return;
    int col = idx & (KK - 1);
    float gamma = normp[0], beta = normp[1];
    h2[idx] = (h2[idx] - stats[col]) * stats[KK + col] * gamma + beta;
}

extern "C" void kernel_launch(void* const* d_in, const int* in_sizes, int n_in,
                              void* d_out, int out_size, void* d_ws, size_t ws_size,
                              hipStream_t stream) {
    const float* nodes  = (const float*)d_in[0];
    const float* rels   = (const float*)d_in[1];
    const float* W1     = (const float*)d_in[2];
    const float* b1     = (const float*)d_in[3];
    const float* W2     = (const float*)d_in[4];
    const float* b2     = (const float*)d_in[5];
    const float* normp  = (const float*)d_in[6];
    const int*   froms  = (const int*)d_in[7];
    const int*   tos    = (const int*)d_in[8];
    const int*   relidx = (const int*)d_in[9];
    const int E = in_sizes[7];
    float* out = (float*)d_out;

    char* ws = (char*)d_ws;
    size_t o = 0;
    auto alloc = [&](size_t bytes) {
        size_t r = o;
        o += (bytes + 255) & ~(size_t)255;
        return r;
    };
    float*    att    = (float*)(ws + alloc((size_t)E * 4));
    unsigned* rowmax = (unsigned*)(ws + alloc((size_t)NR * 4));
    float*    denom  = (float*)(ws + alloc((size_t)NR * 4));
    float*    agg    = (float*)(ws + alloc((size_t)NR * KK * 4));
    float*    h1     = (float*)(ws + alloc((size_t)NN * H1 * 4));
    float*    stats  = (float*)(ws + alloc(2 * KK * 4));

    hipMemsetAsync(rowmax, 0, (size_t)NR * 4, stream);  // enc identity (< enc(-FLT_MAX))
    hipMemsetAsync(denom,  0, (size_t)NR * 4, stream);
    hipMemsetAsync(agg,    0, (size_t)NR * KK * 4, stream);

    att_kernel<<<(E + 7) / 8, 256, 0, stream>>>(nodes, rels, froms, tos, relidx,
                                                att, rowmax, E);
    exp_kernel<<<(E + 255) / 256, 256, 0, stream>>>(froms, relidx, att, rowmax,
                                                    denom, E);
    scatter_kernel<<<(E + 7) / 8, 256, 0, stream>>>(nodes, froms, tos, relidx,
                                                    att, denom, agg, E);
    gemm1_kernel<<<NN / 16, 256, 0, stream>>>(agg, W1, b1, h1);
    gemm2_kernel<<<NN / 16, 256, 0, stream>>>(h1, W2, b2, out);
    stats_kernel<<<KK, 256, 0, stream>>>(out, stats);
    norm_kernel<<<(NN * KK + 255) / 256, 256, 0, stream>>>(out, stats, normp);
}